// SPConvVoxelization_970662609067
// MI455X (gfx1250) — compile-verified
//
#include <hip/hip_runtime.h>
#include <hip/hip_bf16.h>
#include <stdint.h>

// ---------------- Problem constants (from reference) ----------------
#define GX 1440
#define GY 1440
#define GZ 40
#define MAX_PTS 10
#define MAX_VOX 160000
#define PT_C 5

// Hash table: 4M entries, open addressing (linear probe).
// keys+minidx+rank = 48 MB -> resident in MI455X's 192 MB L2, so the
// random-access probe traffic stays on-die (this is the whole design).
#define HT_BITS 22
#define HT_SIZE (1u << HT_BITS)
#define HT_MASK (HT_SIZE - 1u)
#define EMPTY_KEY (-1)
#define INT_MAX_SENTINEL 0x7FFFFFFF

// ---------------- Kernels ----------------

// Zero the entire output buffer (voxels + coords + num_points), int4-wide.
__global__ void vx_zero_out(int* __restrict__ out, long long n) {
    long long n4 = n >> 2;
    int4 z = make_int4(0, 0, 0, 0);
    int4* o4 = reinterpret_cast<int4*>(out);
    for (long long i = (long long)blockIdx.x * blockDim.x + threadIdx.x; i < n4;
         i += (long long)gridDim.x * blockDim.x)
        o4[i] = z;
    if (blockIdx.x == 0 && threadIdx.x < (n & 3)) out[(n4 << 2) + threadIdx.x] = 0;
}

// Init hash table.
__global__ void vx_init_ht(int* __restrict__ ht_key, int* __restrict__ ht_min) {
    unsigned e = blockIdx.x * blockDim.x + threadIdx.x;
    if (e < HT_SIZE) {
        ht_key[e] = EMPTY_KEY;
        ht_min[e] = INT_MAX_SENTINEL;
    }
}

__global__ void vx_zero_words(unsigned* __restrict__ flags, int W) {
    int w = blockIdx.x * blockDim.x + threadIdx.x;
    if (w < W) flags[w] = 0u;
}

// Pass 1: bin points, claim hash entries, record min (first) point index per voxel.
__global__ void vx_pass1(const float* __restrict__ pts, int N,
                         int* __restrict__ ht_key, int* __restrict__ ht_min,
                         int* __restrict__ pt_entry) {
    int i = blockIdx.x * blockDim.x + threadIdx.x;
    if (i >= N) return;
    const float* p = pts + (size_t)i * PT_C;
    // gfx1250 global_prefetch_b8: pull the stream ~40KB ahead into cache.
    __builtin_prefetch(p + PT_C * 2048, 0, 1);
    float x = p[0], y = p[1], z = p[2];
    // Match reference math exactly: floor((p - pmin) / voxel_size), fp32 divide.
    int cx = (int)floorf((x + 54.0f) / 0.075f);
    int cy = (int)floorf((y + 54.0f) / 0.075f);
    int cz = (int)floorf((z + 5.0f) / 0.2f);
    if ((unsigned)cx >= (unsigned)GX || (unsigned)cy >= (unsigned)GY ||
        (unsigned)cz >= (unsigned)GZ) {
        pt_entry[i] = -1;
        return;
    }
    unsigned lin = ((unsigned)cz * GY + (unsigned)cy) * GX + (unsigned)cx;  // < 82.9M, fits i32
    unsigned h = (lin * 2654435761u) & HT_MASK;
    while (true) {
        int old = atomicCAS(&ht_key[h], EMPTY_KEY, (int)lin);
        if (old == EMPTY_KEY || old == (int)lin) {
            atomicMin(&ht_min[h], i);
            pt_entry[i] = (int)h;
            return;
        }
        h = (h + 1u) & HT_MASK;
    }
}

// Mark bit[first_index] for every occupied voxel (first indices are distinct).
__global__ void vx_mark_first(const int* __restrict__ ht_key,
                              const int* __restrict__ ht_min,
                              unsigned* __restrict__ flags) {
    unsigned e = blockIdx.x * blockDim.x + threadIdx.x;
    if (e >= HT_SIZE) return;
    if (ht_key[e] == EMPTY_KEY) return;
    int m = ht_min[e];
    atomicOr(&flags[(unsigned)m >> 5], 1u << ((unsigned)m & 31u));
}

// Stage one flag word per lane from global directly into LDS via the CDNA5
// async path (GLOBAL_LOAD_ASYNC_TO_LDS_B32, tracked with ASYNCcnt). `sh` is
// the kernel's only __shared__ object, so its LDS byte offset is 0 and the
// per-lane LDS destination address is simply tid*4. All 32 lanes are active
// when the op issues (OOB lanes clamp the global address to a valid word).
__device__ __forceinline__ void vx_async_stage_word(const unsigned* __restrict__ flags,
                                                    int w, int W) {
    unsigned widx = (w < W) ? (unsigned)w : (unsigned)(W - 1);
    const unsigned* gaddr = flags + widx;
    unsigned lds_off = threadIdx.x * 4u;  // byte offset of sh[tid] in LDS
    asm volatile("global_load_async_to_lds_b32 %0, %1, off"
                 :
                 : "v"(lds_off), "v"(gaddr)
                 : "memory");
    asm volatile("s_wait_asynccnt 0" ::: "memory");
}

// Hierarchical popcount scan: per-block (256 words) sums.
__global__ void vx_block_sums(const unsigned* __restrict__ flags, int W,
                              int* __restrict__ blockSums) {
    __shared__ int sh[256];
    int w = blockIdx.x * 256 + threadIdx.x;
    vx_async_stage_word(flags, w, W);
    __syncthreads();
    unsigned word = (unsigned)sh[threadIdx.x];
    int v = (w < W) ? __popc(word) : 0;
    __syncthreads();
    sh[threadIdx.x] = v;
    __syncthreads();
    for (int off = 128; off > 0; off >>= 1) {
        if ((int)threadIdx.x < off) sh[threadIdx.x] += sh[threadIdx.x + off];
        __syncthreads();
    }
    if (threadIdx.x == 0) blockSums[blockIdx.x] = sh[0];
}

// Tiny exclusive scan of block sums (B ~ 245), single lane, deterministic.
__global__ void vx_scan_block_sums(int* __restrict__ blockSums, int B) {
    if (blockIdx.x == 0 && threadIdx.x == 0) {
        int acc = 0;
        for (int b = 0; b < B; ++b) {
            int v = blockSums[b];
            blockSums[b] = acc;
            acc += v;
        }
    }
}

// Exclusive per-word prefix = blockOffset + intra-block exclusive scan (LDS).
__global__ void vx_word_prefix(const unsigned* __restrict__ flags, int W,
                               const int* __restrict__ blockOff,
                               int* __restrict__ wordPrefix) {
    __shared__ int sh[256];
    int w = blockIdx.x * 256 + threadIdx.x;
    vx_async_stage_word(flags, w, W);
    __syncthreads();
    unsigned word = (unsigned)sh[threadIdx.x];
    int v = (w < W) ? __popc(word) : 0;
    __syncthreads();
    sh[threadIdx.x] = v;
    __syncthreads();
    for (int off = 1; off < 256; off <<= 1) {
        int t = ((int)threadIdx.x >= off) ? sh[threadIdx.x - off] : 0;
        __syncthreads();
        sh[threadIdx.x] += t;
        __syncthreads();
    }
    if (w < W) wordPrefix[w] = blockOff[blockIdx.x] + sh[threadIdx.x] - v;
}

// Rank each occupied voxel by first-point index; emit coords (z,y,x) for rank < MAX_VOX.
__global__ void vx_assign_rank(const int* __restrict__ ht_key,
                               const int* __restrict__ ht_min,
                               int* __restrict__ ht_rank,
                               const unsigned* __restrict__ flags,
                               const int* __restrict__ wordPrefix,
                               int* __restrict__ coords) {
    unsigned e = blockIdx.x * blockDim.x + threadIdx.x;
    if (e >= HT_SIZE) return;
    int k = ht_key[e];
    if (k == EMPTY_KEY) return;
    unsigned m = (unsigned)ht_min[e];
    unsigned wrd = m >> 5, bit = m & 31u;
    unsigned mask = (bit == 0) ? 0u : (0xFFFFFFFFu >> (32u - bit));
    int r = wordPrefix[wrd] + __popc(flags[wrd] & mask);
    ht_rank[e] = r;
    if (r < MAX_VOX) {
        unsigned lin = (unsigned)k;
        int x = (int)(lin % GX);
        lin /= GX;
        int y = (int)(lin % GY);
        int z = (int)(lin / GY);
        coords[r * 3 + 0] = z;
        coords[r * 3 + 1] = y;
        coords[r * 3 + 2] = x;
    }
}

// Pass 2: scatter point features into voxels, count points per voxel.
__global__ void vx_pass2(const float* __restrict__ pts, int N,
                         const int* __restrict__ pt_entry,
                         const int* __restrict__ ht_rank,
                         float* __restrict__ voxels, int* __restrict__ numpts) {
    int i = blockIdx.x * blockDim.x + threadIdx.x;
    if (i >= N) return;
    __builtin_prefetch(pts + (size_t)(i + 2048) * PT_C, 0, 1);
    int e = pt_entry[i];
    if (e < 0) return;
    int r = ht_rank[e];
    if (r >= MAX_VOX) return;
    int slot = atomicAdd(&numpts[r], 1);
    if (slot < MAX_PTS) {
        const float* p = pts + (size_t)i * PT_C;
        float* d = voxels + ((size_t)r * MAX_PTS + (size_t)slot) * PT_C;
        d[0] = p[0];
        d[1] = p[1];
        d[2] = p[2];
        d[3] = p[3];
        d[4] = p[4];
    }
}

__global__ void vx_clamp_counts(int* __restrict__ numpts) {
    int r = blockIdx.x * blockDim.x + threadIdx.x;
    if (r < MAX_VOX) numpts[r] = min(numpts[r], MAX_PTS);
}

// ---------------- Host launch ----------------
extern "C" void kernel_launch(void* const* d_in, const int* in_sizes, int n_in,
                              void* d_out, int out_size, void* d_ws, size_t ws_size,
                              hipStream_t stream) {
    const float* points = (const float*)d_in[0];
    const int N = in_sizes[0] / PT_C;  // 2,000,000

    // Output layout (flat, reference return order):
    //   voxels:     MAX_VOX*MAX_PTS*5 floats
    //   coords:     MAX_VOX*3 int32 (bit-stored)
    //   num_points: MAX_VOX int32 (bit-stored)
    float* voxels = (float*)d_out;
    int* coords = (int*)(voxels + (size_t)MAX_VOX * MAX_PTS * PT_C);
    int* numpts = coords + (size_t)MAX_VOX * 3;

    // Workspace layout (all 4B aligned):
    char* w = (char*)d_ws;
    int* ht_key = (int*)w;            w += (size_t)HT_SIZE * 4;
    int* ht_min = (int*)w;            w += (size_t)HT_SIZE * 4;
    int* ht_rank = (int*)w;           w += (size_t)HT_SIZE * 4;
    int* pt_entry = (int*)w;          w += (size_t)N * 4;
    const int W = (N + 31) / 32;      // flag words over point indices
    unsigned* flags = (unsigned*)w;   w += (size_t)W * 4;
    int* wordPrefix = (int*)w;        w += (size_t)W * 4;
    const int B = (W + 255) / 256;    // scan blocks
    int* blockSums = (int*)w;         w += (size_t)B * 4;

    const int TB = 256;  // 8 wave32 waves per block
    const long long out_n = (long long)out_size;

    vx_zero_out<<<2048, TB, 0, stream>>>((int*)d_out, out_n);
    vx_init_ht<<<HT_SIZE / TB, TB, 0, stream>>>(ht_key, ht_min);
    vx_zero_words<<<(W + TB - 1) / TB, TB, 0, stream>>>(flags, W);

    vx_pass1<<<(N + TB - 1) / TB, TB, 0, stream>>>(points, N, ht_key, ht_min, pt_entry);
    vx_mark_first<<<HT_SIZE / TB, TB, 0, stream>>>(ht_key, ht_min, flags);

    vx_block_sums<<<B, 256, 0, stream>>>(flags, W, blockSums);
    vx_scan_block_sums<<<1, 32, 0, stream>>>(blockSums, B);
    vx_word_prefix<<<B, 256, 0, stream>>>(flags, W, blockSums, wordPrefix);

    vx_assign_rank<<<HT_SIZE / TB, TB, 0, stream>>>(ht_key, ht_min, ht_rank, flags,
                                                    wordPrefix, coords);
    vx_pass2<<<(N + TB - 1) / TB, TB, 0, stream>>>(points, N, pt_entry, ht_rank,
                                                   voxels, numpts);
    vx_clamp_counts<<<(MAX_VOX + TB - 1) / TB, TB, 0, stream>>>(numpts);
}